// GatherMatMul_24137716203790
// MI455X (gfx1250) — compile-verified
//
#include <hip/hip_runtime.h>

typedef __attribute__((ext_vector_type(2))) float v2f;
typedef __attribute__((ext_vector_type(4))) float v4f;
typedef __attribute__((ext_vector_type(8))) float v8f;

namespace {
constexpr int kD = 4096;        // model dim
constexpr int kB = 16;          // batch
constexpr int kColsPerWave = 64;   // 4 interleaved 16x16 WMMA tiles
constexpr int kWavesPerWG  = 8;    // 256 threads
constexpr int kColsPerWG   = kColsPerWave * kWavesPerWG; // 512
}

// One block: (sample b, 512-column chunk nc, k-chunk kc).
// Each wave computes 64 output columns via 4 accumulator tiles whose columns
// are interleaved (tile t holds columns n_base + 4*l + t, l = lane&15) so that
// every global load of W is a full float4 (b128) from one row.
// KC (split-K factor) is a compile-time constant so klen/strides fold to
// immediates and no integer-division sequence is emitted.
template <int KC>
__global__ __launch_bounds__(256) void gather_gemv_wmma(
    const float* __restrict__ x,       // (B, D)
    const int*   __restrict__ indices, // (B)
    const float* __restrict__ W,       // (8, D, D)
    float*       __restrict__ out)     // (KC, B, D)
{
  constexpr int klen = kD / KC;        // 1024 when KC==4
  __shared__ float xs[klen];           // stage this block's k-chunk of x

  const int b    = blockIdx.x;
  const int nc   = blockIdx.y;
  const int kc   = blockIdx.z;
  const int tid  = threadIdx.x;
  const int wave = tid >> 5;
  const int lane = tid & 31;
  const int l    = lane & 15;  // column slot / N index inside a tile
  const int h    = lane >> 4;  // half-wave selects K rows {2h, 2h+1}

  const int k0 = kc * klen;

  // ---- stage x[b, k0 : k0+klen] into LDS (vectorized, coalesced) ----
  {
    const v4f* __restrict__ x4 = (const v4f*)(x + (size_t)b * kD + k0);
    v4f* xs4 = (v4f*)xs;
#pragma unroll
    for (int i = 0; i < klen / 4; i += 256) xs4[i + tid] = x4[i + tid];
  }
  __syncthreads();

  const int e = indices[b] & 7;
  const float* __restrict__ Wp = W + (size_t)e * kD * kD;

  const int n_base = nc * kColsPerWG + wave * kColsPerWave;
  const int col    = n_base + 4 * l;

  // rows k0+2h, k0+2h+1 of the expert matrix, at this lane's 4 columns
  const float* __restrict__ rowp = Wp + (size_t)(k0 + 2 * h) * kD + col;
  const v2f* __restrict__ xs2 = (const v2f*)xs;

  v8f acc0 = {}, acc1 = {}, acc2 = {}, acc3 = {};

#pragma unroll 4
  for (int kk = 0; kk < klen; kk += 4) {
    // B fragments: two consecutive K rows, 4 contiguous columns (b128 loads)
    v4f w0 = *(const v4f*)rowp;          // row k0+kk+2h
    v4f w1 = *(const v4f*)(rowp + kD);   // row k0+kk+2h+1
    // software prefetch ~16 rows ahead of the streaming reads
    if ((kk & 15) == 0) __builtin_prefetch(rowp + (size_t)16 * kD, 0, 0);
    rowp += (size_t)4 * kD;

    // A fragment: x values broadcast to all 16 M rows (ds_load_b64)
    v2f a = xs2[(kk >> 1) + h];          // { x[kk+2h], x[kk+2h+1] }

    v2f b0 = {w0.x, w1.x};
    v2f b1 = {w0.y, w1.y};
    v2f b2 = {w0.z, w1.z};
    v2f b3 = {w0.w, w1.w};

    acc0 = __builtin_amdgcn_wmma_f32_16x16x4_f32(false, a, false, b0,
                                                 (short)0, acc0, false, false);
    acc1 = __builtin_amdgcn_wmma_f32_16x16x4_f32(false, a, false, b1,
                                                 (short)0, acc1, false, false);
    acc2 = __builtin_amdgcn_wmma_f32_16x16x4_f32(false, a, false, b2,
                                                 (short)0, acc2, false, false);
    acc3 = __builtin_amdgcn_wmma_f32_16x16x4_f32(false, a, false, b3,
                                                 (short)0, acc3, false, false);
  }

  // All 16 M rows of each accumulator are identical (broadcast A); lane group
  // 0-15 holds N=l in C VGPR0. Store 4 interleaved columns as one float4.
  if (h == 0) {
    float* __restrict__ op = out + ((size_t)kc * kB + b) * kD;
    v4f r = {acc0[0], acc1[0], acc2[0], acc3[0]};
    *(v4f*)(op + col) = r;
  }
}

// Deterministic fixed-order reduction of the 4 split-K partials.
__global__ __launch_bounds__(256) void reduce_partials(
    const float* __restrict__ p, float* __restrict__ out)
{
  const int i = blockIdx.x * 256 + threadIdx.x;
  const int n = kB * kD;
  out[i] = (p[i] + p[i + n]) + (p[i + 2 * n] + p[i + 3 * n]);
}

extern "C" void kernel_launch(void* const* d_in, const int* in_sizes, int n_in,
                              void* d_out, int out_size, void* d_ws, size_t ws_size,
                              hipStream_t stream) {
  (void)in_sizes; (void)n_in; (void)out_size;
  const float* x   = (const float*)d_in[0];
  const int*   idx = (const int*)d_in[1];
  const float* W   = (const float*)d_in[2];
  float*       out = (float*)d_out;

  const size_t need = (size_t)4 * kB * kD * sizeof(float);
  if (ws_size >= need) {
    // split-K = 4 -> 16 * 8 * 4 = 512 WGs = 4096 waves for latency hiding
    dim3 grid(kB, kD / kColsPerWG, 4);
    gather_gemv_wmma<4><<<grid, dim3(256), 0, stream>>>(x, idx, W, (float*)d_ws);
    reduce_partials<<<dim3((kB * kD) / 256), dim3(256), 0, stream>>>(
        (const float*)d_ws, out);
  } else {
    dim3 grid(kB, kD / kColsPerWG, 1);
    gather_gemv_wmma<1><<<grid, dim3(256), 0, stream>>>(x, idx, W, out);
  }
}